// GlobalAttention_58213986730699
// MI455X (gfx1250) — compile-verified
//
#include <hip/hip_runtime.h>
#include <math.h>

#define DDIM 1024
#define BB 8
#define TT 50
#define SS 512

typedef __attribute__((ext_vector_type(2))) float v2f;
typedef __attribute__((ext_vector_type(8))) float v8f;

// CDNA5 has a hardware TANH transcendental (ISA transcendental list).
#if __has_builtin(__builtin_amdgcn_tanhf)
#define FAST_TANH(x) __builtin_amdgcn_tanhf(x)
#else
#define FAST_TANH(x) tanhf(x)
#endif

// ---------------------------------------------------------------------------
// enc = tanh(enc_out + coverage * Wcov)   (float4 vectorized)
// ---------------------------------------------------------------------------
__global__ __launch_bounds__(256) void enc_tanh_kernel(
    const float* __restrict__ enc_out, const float* __restrict__ cov,
    const float* __restrict__ Wcov, float* __restrict__ enc, int total4)
{
    int i4 = blockIdx.x * 256 + threadIdx.x;
    if (i4 >= total4) return;
    int base = i4 << 2;
    int bs = base >> 10;            // DDIM == 1024
    int d4 = (base & (DDIM - 1)) >> 2;
    float  c = cov[bs];
    float4 e = ((const float4*)enc_out)[i4];
    float4 w = ((const float4*)Wcov)[d4];
    float4 o;
    o.x = FAST_TANH(fmaf(c, w.x, e.x));
    o.y = FAST_TANH(fmaf(c, w.y, e.y));
    o.z = FAST_TANH(fmaf(c, w.z, e.z));
    o.w = FAST_TANH(fmaf(c, w.w, e.w));
    ((float4*)enc)[i4] = o;
}

// ---------------------------------------------------------------------------
// concat[:, D:2D] = src   (float4 vectorized)
// ---------------------------------------------------------------------------
__global__ __launch_bounds__(256) void concat_src_kernel(
    const float* __restrict__ src, float* __restrict__ concat, int total4)
{
    int i4 = blockIdx.x * 256 + threadIdx.x;
    if (i4 >= total4) return;
    int base = i4 << 2;
    int m = base >> 10;
    int d = base & (DDIM - 1);
    *(float4*)(concat + (long)m * (2 * DDIM) + DDIM + d) = ((const float4*)src)[i4];
}

// ---------------------------------------------------------------------------
// Generic f32 WMMA GEMM:  C[M,N] = A[M,K] * B(^T) + bias
//   BT=true : B is (N,K) row-major (weight)
//   BT=false: B is (K,N) row-major (activation)
// Block = 256 threads = 8 waves; block tile 64x128; K chunked by 32 via LDS.
// Each wave owns a 16x64 strip = FOUR independent 16x16 accumulators
// (4 WMMA chains hide V_WMMA_F32_16X16X4_F32 latency; 1 A-frag feeds 4 WMMAs).
// ---------------------------------------------------------------------------
template<bool BT, bool HAS_BIAS>
__global__ __launch_bounds__(256) void wmma_gemm_kernel(
    const float* __restrict__ A, const float* __restrict__ B,
    const float* __restrict__ bias, float* __restrict__ C,
    int M, int N, int K, int lda, int ldb, int ldc,
    long sA, long sB, long sC)
{
    A += (long)blockIdx.z * sA;
    B += (long)blockIdx.z * sB;
    C += (long)blockIdx.z * sC;

    __shared__ float As[64][33];    // [m][k]
    __shared__ float Bs[128][33];   // [n][k]

    const int tid  = threadIdx.x;
    const int lane = tid & 31;
    const int wave = tid >> 5;     // 0..7
    const int wm   = wave >> 1;    // 0..3 : row tile
    const int wn   = wave & 1;     // 0..1 : 64-wide col strip
    const int m0   = blockIdx.y * 64;
    const int n0   = blockIdx.x * 128;
    const int lr   = lane & 15;            // M (A) / N (B) index within tile
    const int kb   = (lane >> 4) << 1;     // K pair: 0 or 2

    v8f c0 = {}, c1 = {}, c2 = {}, c3 = {};

    for (int kc = 0; kc < K; kc += 32) {
        // ---- stage A tile: 64 rows x 32 k ----
        #pragma unroll
        for (int it = 0; it < 2; ++it) {
            int row = (tid >> 3) + it * 32;
            int col = (tid & 7) << 2;
            int gm  = m0 + row;
            float4 val = make_float4(0.f, 0.f, 0.f, 0.f);
            if (gm < M) val = *(const float4*)(A + (long)gm * lda + (kc + col));
            As[row][col] = val.x; As[row][col + 1] = val.y;
            As[row][col + 2] = val.z; As[row][col + 3] = val.w;
        }
        // ---- stage B tile into Bs[n][k] (128 x 32) ----
        if (BT) {
            #pragma unroll
            for (int it = 0; it < 4; ++it) {
                int row = (tid >> 3) + it * 32;
                int col = (tid & 7) << 2;
                int gn  = n0 + row;
                float4 val = make_float4(0.f, 0.f, 0.f, 0.f);
                if (gn < N) val = *(const float4*)(B + (long)gn * ldb + (kc + col));
                Bs[row][col] = val.x; Bs[row][col + 1] = val.y;
                Bs[row][col + 2] = val.z; Bs[row][col + 3] = val.w;
            }
        } else {
            #pragma unroll
            for (int it = 0; it < 4; ++it) {
                int krow = (tid >> 5) + it * 8;    // 0..31
                int col  = (tid & 31) << 2;        // 0..124
                int gn   = n0 + col;
                float4 val = make_float4(0.f, 0.f, 0.f, 0.f);
                if (gn < N) val = *(const float4*)(B + (long)(kc + krow) * ldb + gn);
                Bs[col][krow]     = val.x;
                Bs[col + 1][krow] = val.y;
                Bs[col + 2][krow] = val.z;
                Bs[col + 3][krow] = val.w;
            }
        }
        // ---- prefetch next K-chunk (global_prefetch_b8) ----
        if (kc + 32 < K) {
            int row = tid >> 3;
            int col = (tid & 7) << 2;
            int gm  = m0 + row;
            if (gm < M) __builtin_prefetch(A + (long)gm * lda + (kc + 32 + col), 0, 3);
            if (BT) {
                int gn = n0 + row;
                if (gn < N) __builtin_prefetch(B + (long)gn * ldb + (kc + 32 + col), 0, 3);
            } else {
                int krow = tid >> 5;
                int c2i  = (tid & 31) << 2;
                int gn   = n0 + c2i;
                if (gn < N) __builtin_prefetch(B + (long)(kc + 32 + krow) * ldb + gn, 0, 3);
            }
        }
        __syncthreads();

        // ---- 32 WMMAs per K-chunk; 4 independent accumulator chains ----
        #pragma unroll
        for (int kk = 0; kk < 32; kk += 4) {
            v2f a, b0, b1, b2, b3;
            a[0]  = As[wm * 16 + lr][kk + kb];
            a[1]  = As[wm * 16 + lr][kk + kb + 1];
            b0[0] = Bs[wn * 64 + lr][kk + kb];
            b0[1] = Bs[wn * 64 + lr][kk + kb + 1];
            b1[0] = Bs[wn * 64 + 16 + lr][kk + kb];
            b1[1] = Bs[wn * 64 + 16 + lr][kk + kb + 1];
            b2[0] = Bs[wn * 64 + 32 + lr][kk + kb];
            b2[1] = Bs[wn * 64 + 32 + lr][kk + kb + 1];
            b3[0] = Bs[wn * 64 + 48 + lr][kk + kb];
            b3[1] = Bs[wn * 64 + 48 + lr][kk + kb + 1];
            c0 = __builtin_amdgcn_wmma_f32_16x16x4_f32(false, a, false, b0, (short)0, c0, false, false);
            c1 = __builtin_amdgcn_wmma_f32_16x16x4_f32(false, a, false, b1, (short)0, c1, false, false);
            c2 = __builtin_amdgcn_wmma_f32_16x16x4_f32(false, a, false, b2, (short)0, c2, false, false);
            c3 = __builtin_amdgcn_wmma_f32_16x16x4_f32(false, a, false, b3, (short)0, c3, false, false);
        }
        __syncthreads();
    }

    // ---- store: C/D layout, VGPR r = row r (lanes 0-15), r+8 (lanes 16-31) ----
    const int cn  = lr;
    const int rof = (lane >> 4) << 3;
    #pragma unroll
    for (int r = 0; r < 8; ++r) {
        int gm = m0 + wm * 16 + r + rof;
        if (gm < M) {
            int gn = n0 + wn * 64 + cn;
            float o0 = c0[r], o1 = c1[r], o2 = c2[r], o3 = c3[r];
            if (HAS_BIAS) {
                o0 += bias[gn];      o1 += bias[gn + 16];
                o2 += bias[gn + 32]; o3 += bias[gn + 48];
            }
            C[(long)gm * ldc + gn]      = o0;
            C[(long)gm * ldc + gn + 16] = o1;
            C[(long)gm * ldc + gn + 32] = o2;
            C[(long)gm * ldc + gn + 48] = o3;
        }
    }
}

// ---------------------------------------------------------------------------
// align[b,t,s] = softmax_s( mask ? -inf : sum_d v[d]*tanh(wq[b,t,d]+uh[b,s,d]) )
// One block per (b,t). 8 waves; one s per wave per pass; each lane streams a
// float4 of uh (b128, 512B per wave-iter) + LDS float4 reads; v_tanh_f32.
// ---------------------------------------------------------------------------
__global__ __launch_bounds__(256) void align_softmax_kernel(
    const float* __restrict__ wq, const float* __restrict__ uh,
    const float* __restrict__ v, const unsigned char* __restrict__ mask,
    float* __restrict__ align)
{
    __shared__ float s_wq[DDIM];
    __shared__ float s_v[DDIM];
    __shared__ float s_sc[SS];
    __shared__ float s_red[8];

    const int bt   = blockIdx.x;
    const int b    = bt / TT;
    const int tid  = threadIdx.x;
    const int lane = tid & 31;
    const int wave = tid >> 5;

    for (int d = tid; d < DDIM; d += 256) {
        s_wq[d] = wq[(long)bt * DDIM + d];
        s_v[d]  = v[d];
    }
    __syncthreads();

    const float4* wq4 = (const float4*)s_wq;
    const float4* v4  = (const float4*)s_v;
    const float* uh_b = uh + (long)b * SS * DDIM;

    for (int s = wave; s < SS; s += 8) {
        const float* row = uh_b + (long)s * DDIM;
        float acc = 0.f;
        #pragma unroll
        for (int j = 0; j < DDIM / 128; ++j) {       // 8 iterations
            int idx4 = j * 32 + lane;
            float4 u  = ((const float4*)row)[idx4];
            float4 wv = wq4[idx4];
            float4 vv = v4[idx4];
            acc = fmaf(vv.x, FAST_TANH(wv.x + u.x), acc);
            acc = fmaf(vv.y, FAST_TANH(wv.y + u.y), acc);
            acc = fmaf(vv.z, FAST_TANH(wv.z + u.z), acc);
            acc = fmaf(vv.w, FAST_TANH(wv.w + u.w), acc);
        }
        #pragma unroll
        for (int off = 16; off > 0; off >>= 1)
            acc += __shfl_xor(acc, off, 32);
        if (lane == 0)
            s_sc[s] = mask[(long)b * SS + s] ? -INFINITY : acc;
    }
    __syncthreads();

    // softmax over S
    float m = -INFINITY;
    for (int s = tid; s < SS; s += 256) m = fmaxf(m, s_sc[s]);
    #pragma unroll
    for (int off = 16; off > 0; off >>= 1) m = fmaxf(m, __shfl_xor(m, off, 32));
    if (lane == 0) s_red[wave] = m;
    __syncthreads();
    float gmax = -INFINITY;
    #pragma unroll
    for (int j = 0; j < 8; ++j) gmax = fmaxf(gmax, s_red[j]);
    __syncthreads();

    float lsum = 0.f;
    for (int s = tid; s < SS; s += 256) {
        float e = __expf(s_sc[s] - gmax);
        s_sc[s] = e;
        lsum += e;
    }
    #pragma unroll
    for (int off = 16; off > 0; off >>= 1) lsum += __shfl_xor(lsum, off, 32);
    if (lane == 0) s_red[wave] = lsum;
    __syncthreads();
    float tot = 0.f;
    #pragma unroll
    for (int j = 0; j < 8; ++j) tot += s_red[j];
    float inv = 1.f / tot;
    for (int s = tid; s < SS; s += 256)
        align[(long)bt * SS + s] = s_sc[s] * inv;
}

// ---------------------------------------------------------------------------
extern "C" void kernel_launch(void* const* d_in, const int* in_sizes, int n_in,
                              void* d_out, int out_size, void* d_ws, size_t ws_size,
                              hipStream_t stream)
{
    (void)in_sizes; (void)n_in; (void)out_size; (void)ws_size;

    const float* src     = (const float*)d_in[0];   // (B,T,D)
    const float* enc_out = (const float*)d_in[1];   // (B,S,D)
    const float* cov     = (const float*)d_in[2];   // (B,S)
    const float* Wq      = (const float*)d_in[3];   // (D,D)
    const float* bq      = (const float*)d_in[4];   // (D)
    const float* Wc      = (const float*)d_in[5];   // (D,D)
    const float* v       = (const float*)d_in[6];   // (D)
    const float* Wout    = (const float*)d_in[7];   // (D,2D)
    const float* bout    = (const float*)d_in[8];   // (D)
    const float* Wcov    = (const float*)d_in[9];   // (D)
    const unsigned char* mask = (const unsigned char*)d_in[10]; // (B,1,S) bool

    float* out    = (float*)d_out;
    float* attn_h = out;                            // (B,T,D)
    float* align  = out + (long)BB * TT * DDIM;     // (B,T,S)

    float* ws = (float*)d_ws;
    long off = 0;
    float* enc    = ws + off; off += (long)BB * SS * DDIM;      // 16 MB
    float* wq     = ws + off; off += (long)BB * TT * DDIM;      // 1.6 MB
    float* uh     = ws + off; off += (long)BB * SS * DDIM;      // 16 MB
    float* concat = ws + off;                                   // 3.2 MB  (B*T, 2D)

    // 1) enc = tanh(enc_out + cov*Wcov)
    {
        int total4 = (BB * SS * DDIM) / 4;
        enc_tanh_kernel<<<(total4 + 255) / 256, 256, 0, stream>>>(
            enc_out, cov, Wcov, enc, total4);
    }
    // 2) wq = src @ Wq^T + bq      (M=400, N=1024, K=1024)
    wmma_gemm_kernel<true, true><<<dim3(DDIM / 128, (BB * TT + 63) / 64, 1), 256, 0, stream>>>(
        src, Wq, bq, wq, BB * TT, DDIM, DDIM, DDIM, DDIM, DDIM, 0, 0, 0);
    // 3) uh = enc @ Wc^T           (M=4096, N=1024, K=1024)
    wmma_gemm_kernel<true, false><<<dim3(DDIM / 128, (BB * SS) / 64, 1), 256, 0, stream>>>(
        enc, Wc, nullptr, uh, BB * SS, DDIM, DDIM, DDIM, DDIM, DDIM, 0, 0, 0);
    // 4) align = softmax_s( v . tanh(wq + uh) )
    align_softmax_kernel<<<BB * TT, 256, 0, stream>>>(wq, uh, v, mask, align);
    // 5) context = align @ enc  -> concat[:, :D]   (batched: M=50, N=1024, K=512)
    wmma_gemm_kernel<false, false><<<dim3(DDIM / 128, (TT + 63) / 64, BB), 256, 0, stream>>>(
        align, enc, nullptr, concat, TT, DDIM, SS, SS, DDIM, 2 * DDIM,
        (long)TT * SS, (long)SS * DDIM, (long)TT * 2 * DDIM);
    // 6) concat[:, D:] = src
    {
        int total4 = (BB * TT * DDIM) / 4;
        concat_src_kernel<<<(total4 + 255) / 256, 256, 0, stream>>>(src, concat, total4);
    }
    // 7) attn_h = concat @ Wout^T + bout  (M=400, N=1024, K=2048)
    wmma_gemm_kernel<true, true><<<dim3(DDIM / 128, (BB * TT + 63) / 64, 1), 256, 0, stream>>>(
        concat, Wout, bout, attn_h, BB * TT, DDIM, 2 * DDIM, 2 * DDIM, 2 * DDIM, DDIM, 0, 0, 0);
}